// RecurrentEncoderLabelling_29858612641822
// MI455X (gfx1250) — compile-verified
//
#include <hip/hip_runtime.h>
#include <math.h>

// Dims from the reference
// B=256, HID=512, H=8, D=64, FF=2048, L=6, V=32000

typedef __attribute__((ext_vector_type(2))) float v2f;
typedef __attribute__((ext_vector_type(8))) float v8f;

#define ACT_NONE 0
#define ACT_ELU1 1
#define ACT_RELU 2

// ---------------------------------------------------------------------------
// GEMM: C[256 x N] = act(A[256 x K] @ W[K x N] + bias)
// One wave computes one 16x16 fp32 tile with V_WMMA_F32_16X16X4_F32.
// Block = 256 threads = 8 waves covering 128 consecutive output columns.
// grid = (N/128, 256/16)
// ---------------------------------------------------------------------------
template<int ACT>
__global__ __launch_bounds__(256) void gemm_wmma_f32(
    const float* __restrict__ A, const float* __restrict__ W,
    const float* __restrict__ bias, float* __restrict__ C,
    int K, int N) {
  const int lane = threadIdx.x & 31;
  const int wave = threadIdx.x >> 5;
  const int l15  = lane & 15;
  const int hi   = lane >> 4;           // 0: K+{0,1} half, 1: K+{2,3} half
  const int tileM = blockIdx.y;
  const int tileN = blockIdx.x * 8 + wave;

  // A fragment: lane l15 holds row M = tileM*16+l15; v0/v1 = K pair
  const float* Arow = A + (size_t)(tileM * 16 + l15) * K + 2 * hi;
  // B fragment: lane l15 holds col N = tileN*16+l15; v0/v1 = K pair
  const float* Wcol = W + (size_t)(2 * hi) * N + (tileN * 16 + l15);

  v8f acc = {0.f, 0.f, 0.f, 0.f, 0.f, 0.f, 0.f, 0.f};

  #pragma unroll 8
  for (int k0 = 0; k0 < K; k0 += 4) {
    v2f a = *(const v2f*)(Arow + k0);          // A[row][k0+2*hi .. +1]
    v2f b;
    b.x = Wcol[(size_t)k0 * N];                // W[k0+2*hi  ][col]
    b.y = Wcol[(size_t)k0 * N + N];            // W[k0+2*hi+1][col]
    // 8 args: (neg_a, A, neg_b, B, c_mod, C, reuse_a, reuse_b)
    acc = __builtin_amdgcn_wmma_f32_16x16x4_f32(
        false, a, false, b, (short)0, acc, false, false);
  }

  // D layout: VGPR g -> row g (lanes 0-15) / row 8+g (lanes 16-31), col = l15
  const int col = tileN * 16 + l15;
  const float bcol = bias[col];
  #pragma unroll
  for (int g = 0; g < 8; ++g) {
    const int row = tileM * 16 + hi * 8 + g;
    float x = acc[g] + bcol;
    if (ACT == ACT_ELU1) x = (x > 0.f) ? (x + 1.f) : __expf(x);  // elu(x)+1
    if (ACT == ACT_RELU) x = fmaxf(x, 0.f);
    C[(size_t)row * N + col] = x;
  }
}

// ---------------------------------------------------------------------------
// Fused linear-attention state update. One block per (b,h):
//   Si = S + k (x) v        (S read once, written once -> the 402 MB floor)
//   num[m] = sum_d q[d]*Si[d][m] ;  den = sum_d q[d]*(Z[d]+k[d]) + eps
//   attn[m] = num[m]/den ;  Z_out = Z + k
// 256 threads = 64 columns (m) x 4 d-groups; fully coalesced over m.
// ---------------------------------------------------------------------------
__global__ __launch_bounds__(256) void state_attn(
    const float* __restrict__ q, const float* __restrict__ k,
    const float* __restrict__ v, const float* __restrict__ Sin,
    const float* __restrict__ Zin, float* __restrict__ Sout,
    float* __restrict__ Zout, float* __restrict__ attn) {
  __shared__ float sq[64], sk[64], sv[64];
  __shared__ float red[256], redd[64];
  __shared__ float sden;
  const int t = threadIdx.x;
  const size_t bh = blockIdx.x;

  if (t < 64) {
    sq[t] = q[bh * 64 + t];
    sk[t] = k[bh * 64 + t];
    sv[t] = v[bh * 64 + t];
  }
  __syncthreads();

  const int m  = t & 63;
  const int dg = t >> 6;
  const float* Sb = Sin  + bh * 4096;
  float*       So = Sout + bh * 4096;

  float acc = 0.f;
  #pragma unroll 4
  for (int d = dg; d < 64; d += 4) {
    float s = Sb[d * 64 + m] + sk[d] * sv[m];
    So[d * 64 + m] = s;
    acc = fmaf(sq[d], s, acc);
  }
  red[t] = acc;
  if (t < 64) {
    float zn = Zin[bh * 64 + t] + sk[t];
    Zout[bh * 64 + t] = zn;
    redd[t] = sq[t] * zn;
  }
  __syncthreads();
  if (t == 0) {
    float dsum = 0.f;
    #pragma unroll
    for (int i = 0; i < 64; ++i) dsum += redd[i];
    sden = dsum + 1e-6f;
  }
  __syncthreads();
  if (dg == 0) {
    float num = red[m] + red[m + 64] + red[m + 128] + red[m + 192];
    attn[bh * 64 + m] = num / sden;
  }
}

// ---------------------------------------------------------------------------
// LayerNorm over HID=512 with optional residual: out = LN(h + r)*g + b
// One block of 256 threads per row (2 elems/thread).
// ---------------------------------------------------------------------------
__global__ __launch_bounds__(256) void ln_residual(
    const float* __restrict__ h, const float* __restrict__ r,
    const float* __restrict__ g, const float* __restrict__ b,
    float* __restrict__ out) {
  __shared__ float red[256];
  __shared__ float s_mean, s_rstd;
  const int t = threadIdx.x;
  const size_t base = (size_t)blockIdx.x * 512;

  float x0 = h[base + t];
  float x1 = h[base + t + 256];
  if (r) { x0 += r[base + t]; x1 += r[base + t + 256]; }

  red[t] = x0 + x1;
  __syncthreads();
  for (int s = 128; s > 0; s >>= 1) {
    if (t < s) red[t] += red[t + s];
    __syncthreads();
  }
  if (t == 0) s_mean = red[0] * (1.f / 512.f);
  __syncthreads();
  const float mean = s_mean;
  const float d0 = x0 - mean, d1 = x1 - mean;

  red[t] = d0 * d0 + d1 * d1;
  __syncthreads();
  for (int s = 128; s > 0; s >>= 1) {
    if (t < s) red[t] += red[t + s];
    __syncthreads();
  }
  if (t == 0) s_rstd = rsqrtf(red[0] * (1.f / 512.f) + 1e-5f);
  __syncthreads();
  const float rs = s_rstd;

  out[base + t]       = d0 * rs * g[t]       + b[t];
  out[base + t + 256] = d1 * rs * g[t + 256] + b[t + 256];
}

// ---------------------------------------------------------------------------
// h[b] = emb[x[b]] + sinusoidal_pe(pos). One block per batch row.
// pe[2j] = sin(pos * 10000^(-j/256)), pe[2j+1] = cos(...)
// ---------------------------------------------------------------------------
__global__ __launch_bounds__(256) void embed_pe(
    const int* __restrict__ x, const int* __restrict__ pos_p,
    const float* __restrict__ emb, float* __restrict__ h) {
  const int b = blockIdx.x;
  const int t = threadIdx.x;
  const int tok = x[b];
  const float pos = (float)pos_p[0];
  #pragma unroll
  for (int e = 0; e < 2; ++e) {
    const int i = t + e * 256;
    const int j = i >> 1;
    // 10000^(-j/256) = exp(-ln(10000) * j/256)
    const float ang = pos * __expf(-9.210340371976184f * ((float)j * (1.f / 256.f)));
    const float pe = (i & 1) ? __cosf(ang) : __sinf(ang);
    h[(size_t)b * 512 + i] = emb[(size_t)tok * 512 + i] + pe;
  }
}

// ---------------------------------------------------------------------------
extern "C" void kernel_launch(void* const* d_in, const int* in_sizes, int n_in,
                              void* d_out, int out_size, void* d_ws, size_t ws_size,
                              hipStream_t stream) {
  const int*   x    = (const int*)  d_in[0];
  const int*   pos  = (const int*)  d_in[1];
  const float* S    = (const float*)d_in[2];
  const float* Z    = (const float*)d_in[3];
  const float* emb  = (const float*)d_in[4];
  const float* Wq   = (const float*)d_in[5];
  const float* bq   = (const float*)d_in[6];
  const float* Wk   = (const float*)d_in[7];
  const float* bk   = (const float*)d_in[8];
  const float* Wv   = (const float*)d_in[9];
  const float* bv   = (const float*)d_in[10];
  const float* Wo   = (const float*)d_in[11];
  const float* bo   = (const float*)d_in[12];
  const float* W1   = (const float*)d_in[13];
  const float* b1   = (const float*)d_in[14];
  const float* W2   = (const float*)d_in[15];
  const float* b2   = (const float*)d_in[16];
  const float* ln1g = (const float*)d_in[17];
  const float* ln1b = (const float*)d_in[18];
  const float* ln2g = (const float*)d_in[19];
  const float* ln2b = (const float*)d_in[20];
  const float* lnfg = (const float*)d_in[21];
  const float* lnfb = (const float*)d_in[22];

  // d_out = concat(h[256x512], S_out[6x256x8x64x64], Z_out[6x256x8x64])
  float* out   = (float*)d_out;
  float* h_out = out;
  float* Sout  = out + 131072;
  float* Zout  = out + 131072 + (size_t)50331648;

  // workspace layout (floats)
  float* ws = (float*)d_ws;
  float* h  = ws;                 // [256,512]
  float* qb = ws + 131072;        // [256,512]
  float* kb = ws + 262144;        // [256,512]
  float* vb = ws + 393216;        // [256,512]
  float* ab = ws + 524288;        // attn vec [256,512]
  float* tb = ws + 655360;        // attn @ Wo [256,512]
  float* fb = ws + 786432;        // ffn hidden [256,2048]
  float* yb = ws + 1310720;       // ffn out [256,512]

  const size_t WW  = (size_t)512 * 512;
  const size_t W1S = (size_t)512 * 2048;
  const size_t SBH = (size_t)256 * 8 * 64 * 64;  // per-layer S elems
  const size_t ZBH = (size_t)256 * 8 * 64;       // per-layer Z elems

  dim3 blk(256);
  dim3 g512(4, 16);    // N=512:  4 col-blocks x 16 row-tiles
  dim3 g2048(16, 16);  // N=2048: 16 col-blocks x 16 row-tiles

  embed_pe<<<256, blk, 0, stream>>>(x, pos, emb, h);

  for (int l = 0; l < 6; ++l) {
    gemm_wmma_f32<ACT_ELU1><<<g512, blk, 0, stream>>>(
        h, Wq + l * WW, bq + l * 512, qb, 512, 512);
    gemm_wmma_f32<ACT_ELU1><<<g512, blk, 0, stream>>>(
        h, Wk + l * WW, bk + l * 512, kb, 512, 512);
    gemm_wmma_f32<ACT_NONE><<<g512, blk, 0, stream>>>(
        h, Wv + l * WW, bv + l * 512, vb, 512, 512);

    state_attn<<<2048, blk, 0, stream>>>(
        qb, kb, vb, S + l * SBH, Z + l * ZBH,
        Sout + l * SBH, Zout + l * ZBH, ab);

    gemm_wmma_f32<ACT_NONE><<<g512, blk, 0, stream>>>(
        ab, Wo + l * WW, bo + l * 512, tb, 512, 512);
    ln_residual<<<256, blk, 0, stream>>>(
        h, tb, ln1g + l * 512, ln1b + l * 512, h);

    gemm_wmma_f32<ACT_RELU><<<g2048, blk, 0, stream>>>(
        h, W1 + l * W1S, b1 + l * 2048, fb, 512, 2048);
    gemm_wmma_f32<ACT_NONE><<<g512, blk, 0, stream>>>(
        fb, W2 + l * W1S, b2 + l * 512, yb, 2048, 512);
    ln_residual<<<256, blk, 0, stream>>>(
        h, yb, ln2g + l * 512, ln2b + l * 512, h);
  }

  ln_residual<<<256, blk, 0, stream>>>(h, nullptr, lnfg, lnfb, h_out);
}